// GlobalPointer_63299228008708
// MI455X (gfx1250) — compile-verified
//
#include <hip/hip_runtime.h>
#include <math.h>

// GlobalPointer head for MI455X (gfx1250, wave32).
// Stage 1: X(8192x768) @ W(768x1536) + b  -> RoPE -> Q/K buffers [B][H][L][64] (in d_ws)
// Stage 2: per (b,h): Q(512x64) @ K^T -> mask/tril/scale -> out (nontemporal stores)
// Both GEMMs use V_WMMA_F32_16X16X4_F32 (full fp32, matches reference precision).

typedef __attribute__((ext_vector_type(2))) float v2f;
typedef __attribute__((ext_vector_type(8))) float v8f;

#define BB   16
#define LL   512
#define HIDN 768
#define HH   12
#define DD   64
#define NQK  1536           /* HH * 2 * DD */
#define BIGNEG 1000000000000.0f
#define LN1E4_OVER_32 0.28782313662425572f  /* ln(10000)/32 */

__device__ __forceinline__ v8f wmma_f32(v2f a, v2f b, v8f c) {
  // 8 args: (neg_a, A, neg_b, B, c_mod, C, reuse_a, reuse_b)
  return __builtin_amdgcn_wmma_f32_16x16x4_f32(false, a, false, b, (short)0, c,
                                               false, false);
}

// ---------------------------------------------------------------------------
// Stage 1: projection GEMM + bias + RoPE, scatter into Q / K workspace.
// Wave tile: 16 rows (of 8192) x 128 cols (of 1536). 512*12 = 6144 waves.
// A-frag layout (32-bit A 16x4): lane<16 holds K=kb+0,kb+1; lane>=16 holds kb+2,kb+3.
// ---------------------------------------------------------------------------
__global__ void __launch_bounds__(256)
qk_rope_gemm(const float* __restrict__ X, const float* __restrict__ W,
             const float* __restrict__ bias,
             float* __restrict__ Qb, float* __restrict__ Kb)
{
  const int wave = blockIdx.x * (blockDim.x >> 5) + (threadIdx.x >> 5);
  const int lane = threadIdx.x & 31;
  const int half = lane >> 4;   // 0: lanes 0-15, 1: lanes 16-31
  const int lr   = lane & 15;

  const int NSTRIP = NQK / 128;           // 12 column strips
  const int m0 = (wave / NSTRIP) * 16;    // row block within [0, B*L)
  const int n0 = (wave % NSTRIP) * 128;   // col block within [0, 1536)

  v8f acc[8] = {};

  const float* arow = X + (size_t)(m0 + lr) * HIDN + 2 * half;
  const float* wcol = W + n0 + lr;

  for (int kb = 0; kb < HIDN; kb += 4) {
    v2f a = *(const v2f*)(arow + kb);                       // global_load_b64
    const float* w0 = wcol + (size_t)(kb + 2 * half) * NQK; // B rows kb+2*half, +1
#pragma unroll
    for (int t = 0; t < 8; ++t) {
      v2f bf;
      bf.x = w0[t * 16];
      bf.y = w0[t * 16 + NQK];
      acc[t] = wmma_f32(a, bf, acc[t]);
    }
  }

  const int bidx = m0 / LL;   // batch index (tile never crosses batch: 512 % 16 == 0)

#pragma unroll
  for (int t = 0; t < 8; ++t) {
    const int n  = n0 + t * 16 + lr;      // column in [0,1536)
    const int h  = n >> 7;                // head
    const int c  = n & 127;               // 0..127 within head (q: 0..63, k: 64..127)
    const int d  = c & 63;                // rotary dim
    const bool is_k = (c >= 64);
    const float bn   = bias[n];
    const float invf = __expf(-LN1E4_OVER_32 * (float)(d >> 1)); // 10000^(-(d/2)/32)
    const float sgn  = (d & 1) ? 1.0f : -1.0f; // even d: -x[d+1]; odd d: +x[d-1]
    float* obase = (is_k ? Kb : Qb) + (((size_t)bidx * HH + h) * LL) * DD + d;
#pragma unroll
    for (int r = 0; r < 8; ++r) {
      const int m = m0 + half * 8 + r;    // C/D row: lanes16-31 hold M = r+8
      const int p = m & (LL - 1);         // sequence position
      float v = acc[t][r] + bn;
      // partner lane holds column d^1 for the same row (same half, lr^1)
      float partner = __shfl_xor(v, 1, 32);
      float x2  = sgn * partner;
      float ang = (float)p * invf;
      float cv  = cosf(ang);
      float sv  = sinf(ang);
      obase[(size_t)p * DD] = v * cv + x2 * sv;
    }
  }
}

// ---------------------------------------------------------------------------
// Stage 2: per (b,h) logits = Q @ K^T, then pad/causal mask and 1/8 scale.
// Wave tile: 16 rows x 128 cols of one 512x512 score matrix.
// 192 heads * 32 m-tiles * 4 n-strips = 24576 waves.
// ---------------------------------------------------------------------------
__global__ void __launch_bounds__(256)
logits_gemm(const float* __restrict__ Qb, const float* __restrict__ Kb,
            const float* __restrict__ mask, float* __restrict__ out)
{
  const int wave = blockIdx.x * (blockDim.x >> 5) + (threadIdx.x >> 5);
  const int lane = threadIdx.x & 31;
  const int half = lane >> 4;
  const int lr   = lane & 15;

  const int NSTRIP = LL / 128;            // 4
  const int MTILES = LL / 16;             // 32
  const int per_bh = MTILES * NSTRIP;     // 128
  const int bh  = wave / per_bh;
  const int rem = wave % per_bh;
  const int m0  = (rem / NSTRIP) * 16;
  const int n0  = (rem % NSTRIP) * 128;
  const int b   = bh / HH;

  const float* qbase = Qb + (size_t)bh * LL * DD;
  const float* kbase = Kb + (size_t)bh * LL * DD;

  v8f acc[8] = {};

  const float* aq = qbase + (size_t)(m0 + lr) * DD + 2 * half;

#pragma unroll 4
  for (int kb = 0; kb < DD; kb += 4) {
    v2f a = *(const v2f*)(aq + kb);       // Q rows, contiguous b64
#pragma unroll
    for (int t = 0; t < 8; ++t) {
      // B[k][n] = K[n][k]: lane lr supplies column n0+t*16+lr, rows kb+2*half..+1
      v2f bf = *(const v2f*)(kbase + (size_t)(n0 + t * 16 + lr) * DD + kb + 2 * half);
      acc[t] = wmma_f32(a, bf, acc[t]);
    }
  }

#pragma unroll
  for (int t = 0; t < 8; ++t) {
    const int n = n0 + t * 16 + lr;
    const float pad = mask[b * LL + n];
    const float colterm = -(1.0f - pad) * BIGNEG;
    float* obase = out + ((size_t)bh * LL) * LL + n;
#pragma unroll
    for (int r = 0; r < 8; ++r) {
      const int m = m0 + half * 8 + r;
      float v = acc[t][r] * pad + colterm - ((m > n) ? BIGNEG : 0.0f);
      // streaming 201 MB result: non-temporal so Q/K stay resident in L2
      __builtin_nontemporal_store(v * 0.125f, &obase[(size_t)m * LL]);
    }
  }
}

extern "C" void kernel_launch(void* const* d_in, const int* in_sizes, int n_in,
                              void* d_out, int out_size, void* d_ws, size_t ws_size,
                              hipStream_t stream) {
  const float* X    = (const float*)d_in[0];  // (16,512,768)
  const float* mask = (const float*)d_in[1];  // (16,512)
  const float* W    = (const float*)d_in[2];  // (768,1536)
  const float* bias = (const float*)d_in[3];  // (1536,)
  float* out = (float*)d_out;                 // (16,12,512,512)

  float* Qb = (float*)d_ws;                               // 16*12*512*64 floats
  float* Kb = Qb + (size_t)BB * HH * LL * DD;             // same size (total ~50 MB)

  // Stage 1: (8192/16) * (1536/128) = 6144 waves, 8 waves (256 thr) per block
  qk_rope_gemm<<<6144 / 8, 256, 0, stream>>>(X, W, bias, Qb, Kb);

  // Stage 2: 192 * 32 * 4 = 24576 waves
  logits_gemm<<<24576 / 8, 256, 0, stream>>>(Qb, Kb, mask, out);
}